// HeteroSemanticModel_3805341024339
// MI455X (gfx1250) — compile-verified
//
#include <hip/hip_runtime.h>

#define NP_ 50000
#define NA_ 100000
#define EC_ 300000
#define EW_ 500000
#define PD_ 768
#define AD_ 128
#define ED_ 768
#define HD_ 64

#define ROWPAD 68           // LDS row pitch in floats (64 data + 4 pad) -> bank-conflict-free ds_load_b64
#define CHUNKS (ED_ / 64)   // 12 chunks of K=64

typedef __attribute__((ext_vector_type(2))) float v2f;
typedef __attribute__((ext_vector_type(8))) float v8f;

// ---------------- workspace layout (float offsets) ----------------
#define NP64 ((long)NP_ * 64)   // 3,200,000
#define NA64 ((long)NA_ * 64)   // 6,400,000
#define EC64 ((long)EC_ * 64)   // 19,200,000
#define OFF_XP    0L
#define OFF_XA    (OFF_XP + NP64)
#define OFF_Q     (OFF_XA + NA64)
#define OFF_K     (OFF_Q + NP64)
#define OFF_V     (OFF_K + NP64)
#define OFF_SR    (OFF_V + NP64)
#define OFF_HS    (OFF_SR + NP64)
#define OFF_EF    (OFF_HS + NP64)
#define OFF_AH    (OFF_EF + EC64)
#define OFF_ARAW  (OFF_AH + (long)EC_ * 3)
#define OFF_EX    (OFF_ARAW + EC_)
#define OFF_MMAX  (OFF_EX + EC_)
// zero-init region (contiguous): agg, deg, denom, hgat
#define OFF_AGG   (OFF_MMAX + NP_)
#define OFF_DEG   (OFF_AGG + NP64)
#define OFF_DENOM (OFF_DEG + NP_)
#define OFF_HGAT  (OFF_DENOM + NP_)
#define ZERO_LEN  (NP64 + NP_ + NP_ + NP64)
#define OFF_HP    (OFF_HGAT + NP64)

// ---------------- init ----------------
__global__ __launch_bounds__(256) void init_ws_k(float* zbase, long nz, float* mmax, int np) {
  long gid = (long)blockIdx.x * 256 + threadIdx.x;
  if (gid < nz) zbase[gid] = 0.0f;
  if (gid < np) mmax[gid] = -3.4028235e38f;
}

// ------------- generic GEMM: C[M,64] = act(A[M,K] @ W[K,64] + bias) -------------
// one wave per 16-row M tile; 4 f32 WMMA accumulators cover all 64 columns.
// K strip-mined into 64-wide chunks: one WGP-scope prefetch per chunk, fixed-trip
// unrolled inner loop (K is always a multiple of 64 here: 768 / 128 / 64).
__global__ __launch_bounds__(128) void gemm_n64_k(const float* __restrict__ A,
    const float* __restrict__ W, const float* __restrict__ bias,
    float* __restrict__ C, int Mtiles, int K, int relu)
{
  int wave = threadIdx.x >> 5;
  int lane = threadIdx.x & 31;
  int mtile = blockIdx.x * 4 + wave;
  if (mtile >= Mtiles) return;
  int half = lane >> 4, l = lane & 15;
  const float* arow = A + ((long)mtile * 16 + l) * K;
  v8f a0 = {}, a1 = {}, a2 = {}, a3 = {};
  for (int kc = 0; kc < K; kc += 64) {
    if (kc + 64 < K) __builtin_prefetch(arow + kc + 64, 0, 3);   // WGP-scope, all levels
#pragma unroll 4
    for (int kk = 0; kk < 64; kk += 4) {
      int ka = kc + kk + half * 2;
      v2f a = *(const v2f*)(arow + ka);            // A[row][ka], A[row][ka+1]
      const float* w0 = W + (long)ka * 64;
      const float* w1 = w0 + 64;
      v2f b0, b1, b2, b3;
      b0.x = w0[l];      b0.y = w1[l];
      b1.x = w0[l + 16]; b1.y = w1[l + 16];
      b2.x = w0[l + 32]; b2.y = w1[l + 32];
      b3.x = w0[l + 48]; b3.y = w1[l + 48];
      a0 = __builtin_amdgcn_wmma_f32_16x16x4_f32(false, a, false, b0, (short)0, a0, false, false);
      a1 = __builtin_amdgcn_wmma_f32_16x16x4_f32(false, a, false, b1, (short)0, a1, false, false);
      a2 = __builtin_amdgcn_wmma_f32_16x16x4_f32(false, a, false, b2, (short)0, a2, false, false);
      a3 = __builtin_amdgcn_wmma_f32_16x16x4_f32(false, a, false, b3, (short)0, a3, false, false);
    }
  }
  float bi0 = bias ? bias[l] : 0.f;
  float bi1 = bias ? bias[l + 16] : 0.f;
  float bi2 = bias ? bias[l + 32] : 0.f;
  float bi3 = bias ? bias[l + 48] : 0.f;
  for (int r = 0; r < 8; ++r) {
    long orow = (long)mtile * 16 + r + half * 8;
    float* crow = C + orow * 64;
    float c0 = a0[r] + bi0, c1 = a1[r] + bi1, c2 = a2[r] + bi2, c3 = a3[r] + bi3;
    if (relu) { c0 = fmaxf(c0, 0.f); c1 = fmaxf(c1, 0.f); c2 = fmaxf(c2, 0.f); c3 = fmaxf(c3, 0.f); }
    crow[l] = c0; crow[l + 16] = c1; crow[l + 32] = c2; crow[l + 48] = c3;
  }
}

// ------------- fused: ef = attr@We + be  AND  ah = attr@W_head[128:896] -------------
// Streams the 922 MB attr matrix once, via single-buffered ASYNC loads to LDS
// (coalesced 16B segments, ASYNCcnt-gated), then feeds WMMA fragments from LDS.
__device__ __forceinline__ void issue_chunk_async(const float* abase, unsigned wbase_b,
                                                  int lane, int c) {
#pragma unroll
  for (int i = 0; i < 8; ++i) {
    int seg = i * 32 + lane;                 // 256 x 16B segments = 16 rows x 64 floats
    int row = seg >> 4, cs = seg & 15;
    const float* ga = abase + (long)row * ED_ + c * 64 + cs * 4;
    unsigned lo = wbase_b + (unsigned)(row * ROWPAD + cs * 4) * 4u;
    asm volatile("global_load_async_to_lds_b128 %0, %1, off"
                 :: "v"(lo), "v"(ga) : "memory");
  }
}

__global__ __launch_bounds__(128) void efeat_head_k(const float* __restrict__ attr,
    const float* __restrict__ We, const float* __restrict__ be,
    const float* __restrict__ Wh /* 768 x 3 */,
    float* __restrict__ ef, float* __restrict__ ah, int Mtiles)
{
  extern __shared__ float smem[];            // 4 waves * 16 rows * ROWPAD floats
  int wave = threadIdx.x >> 5;
  int lane = threadIdx.x & 31;
  int mtile = blockIdx.x * 4 + wave;
  if (mtile >= Mtiles) return;
  int half = lane >> 4, l = lane & 15;
  unsigned wbase_b = (unsigned)(wave * 16 * ROWPAD) * 4u;   // byte offset of this wave's buffer
  float* wbuf = smem + (size_t)wave * 16 * ROWPAD;
  const float* abase = attr + (long)mtile * 16 * ED_;

  v8f a0 = {}, a1 = {}, a2 = {}, a3 = {};
  float p0 = 0.f, p1 = 0.f, p2 = 0.f;

  issue_chunk_async(abase, wbase_b, lane, 0);
  for (int c = 0; c < CHUNKS; ++c) {
    if (c + 1 < CHUNKS) {
      issue_chunk_async(abase, wbase_b, lane, c + 1);
      // async loads complete in order: <=8 outstanding => chunk c is resident
      asm volatile("s_wait_asynccnt 0x8" ::: "memory");
    } else {
      asm volatile("s_wait_asynccnt 0x0" ::: "memory");
    }
#pragma unroll 4
    for (int kk = 0; kk < 64; kk += 4) {
      int ka = kk + half * 2;
      int kg = c * 64 + ka;                  // global K index
      v2f a = *(const v2f*)(wbuf + l * ROWPAD + ka);   // ds_load_b64, bank-conflict-free
      const float* w0 = We + (long)kg * 64;
      const float* w1 = w0 + 64;
      v2f b0, b1, b2, b3;
      b0.x = w0[l];      b0.y = w1[l];
      b1.x = w0[l + 16]; b1.y = w1[l + 16];
      b2.x = w0[l + 32]; b2.y = w1[l + 32];
      b3.x = w0[l + 48]; b3.y = w1[l + 48];
      a0 = __builtin_amdgcn_wmma_f32_16x16x4_f32(false, a, false, b0, (short)0, a0, false, false);
      a1 = __builtin_amdgcn_wmma_f32_16x16x4_f32(false, a, false, b1, (short)0, a1, false, false);
      a2 = __builtin_amdgcn_wmma_f32_16x16x4_f32(false, a, false, b2, (short)0, a2, false, false);
      a3 = __builtin_amdgcn_wmma_f32_16x16x4_f32(false, a, false, b3, (short)0, a3, false, false);
      // head partials along same A fragment (lane covers K = kg, kg+1 of row l)
      p0 += a.x * Wh[kg * 3 + 0] + a.y * Wh[(kg + 1) * 3 + 0];
      p1 += a.x * Wh[kg * 3 + 1] + a.y * Wh[(kg + 1) * 3 + 1];
      p2 += a.x * Wh[kg * 3 + 2] + a.y * Wh[(kg + 1) * 3 + 2];
    }
  }

  float bi0 = be[l], bi1 = be[l + 16], bi2 = be[l + 32], bi3 = be[l + 48];
  for (int r = 0; r < 8; ++r) {
    long orow = (long)mtile * 16 + r + half * 8;
    float* crow = ef + orow * 64;
    crow[l] = a0[r] + bi0; crow[l + 16] = a1[r] + bi1;
    crow[l + 32] = a2[r] + bi2; crow[l + 48] = a3[r] + bi3;
  }
  // lane l and lane l+16 hold complementary K halves of row l
  p0 += __shfl_xor(p0, 16);
  p1 += __shfl_xor(p1, 16);
  p2 += __shfl_xor(p2, 16);
  if (half == 0) {
    long row = (long)mtile * 16 + l;
    ah[row * 3 + 0] = p0; ah[row * 3 + 1] = p1; ah[row * 3 + 2] = p2;
  }
}

// ---------------- SAGE mean aggregation ----------------
__global__ __launch_bounds__(256) void sage_scatter_k(const float* __restrict__ xa,
    const int* __restrict__ wsrc, const int* __restrict__ wdst,
    float* __restrict__ agg, float* __restrict__ deg, long total)
{
  long gid = (long)blockIdx.x * 256 + threadIdx.x;
  if (gid >= total) return;
  long e = gid >> 6; int c = (int)(gid & 63);
  int s = wsrc[e], d = wdst[e];
  atomicAdd(&agg[(long)d * 64 + c], xa[(long)s * 64 + c]);
  if (c == 0) atomicAdd(&deg[d], 1.0f);
}

__global__ __launch_bounds__(256) void sage_mean_k(float* __restrict__ agg,
    const float* __restrict__ deg, long n)
{
  long gid = (long)blockIdx.x * 256 + threadIdx.x;
  if (gid >= n) return;
  agg[gid] /= fmaxf(deg[gid >> 6], 1.0f);
}

// ---------------- attention: raw alpha + segment max ----------------
__device__ inline void atomicMaxF(float* addr, float val) {
  if (val >= 0.f) atomicMax((int*)addr, __float_as_int(val));
  else atomicMin((unsigned int*)addr, __float_as_uint(val));
}

__global__ __launch_bounds__(256) void alpha_raw_k(const float* __restrict__ q,
    const float* __restrict__ k, const float* __restrict__ ef,
    const int* __restrict__ src, const int* __restrict__ dst,
    const float* __restrict__ dt, const float* __restrict__ lam,
    float* __restrict__ araw, float* __restrict__ mmax, int EC)
{
  int e = (blockIdx.x * 256 + threadIdx.x) >> 5;
  int lane = threadIdx.x & 31;
  if (e >= EC) return;
  int s = src[e], d = dst[e];
  const float* qd = q + (long)d * 64;
  const float* ks = k + (long)s * 64;
  const float* ee = ef + (long)e * 64;
  float sum = qd[lane] * ks[lane] * ee[lane] + qd[lane + 32] * ks[lane + 32] * ee[lane + 32];
  for (int off = 16; off; off >>= 1) sum += __shfl_xor(sum, off);
  if (lane == 0) {
    float al = sum * 0.125f * __expf(-fabsf(lam[0]) * dt[e]);
    araw[e] = al;
    atomicMaxF(&mmax[d], al);
  }
}

// ---------------- exp, segment-sum, and weighted-V scatter ----------------
__global__ __launch_bounds__(256) void ex_scatter_k(const float* __restrict__ v,
    const float* __restrict__ araw, const float* __restrict__ mmax,
    const int* __restrict__ src, const int* __restrict__ dst,
    float* __restrict__ ex, float* __restrict__ denom, float* __restrict__ hgat, int EC)
{
  int e = (blockIdx.x * 256 + threadIdx.x) >> 5;
  int lane = threadIdx.x & 31;
  if (e >= EC) return;
  int s = src[e], d = dst[e];
  float exv = __expf(araw[e] - mmax[d]);
  if (lane == 0) { ex[e] = exv; atomicAdd(&denom[d], exv); }
  const float* vs = v + (long)s * 64;
  atomicAdd(&hgat[(long)d * 64 + lane], exv * vs[lane]);
  atomicAdd(&hgat[(long)d * 64 + lane + 32], exv * vs[lane + 32]);
}

// ---------------- h_papers = relu(hs + sr + hgat/denom) ----------------
__global__ __launch_bounds__(256) void finalize_k(const float* __restrict__ hs,
    const float* __restrict__ sr, const float* __restrict__ hgat,
    const float* __restrict__ denom, float* __restrict__ hp, long n)
{
  long gid = (long)blockIdx.x * 256 + threadIdx.x;
  if (gid >= n) return;
  float g = hgat[gid] / (denom[gid >> 6] + 1e-16f);
  hp[gid] = fmaxf(hs[gid] + sr[gid] + g, 0.f);
}

// ---------------- fused intent head + alpha output ----------------
__global__ __launch_bounds__(256) void head_k(const float* __restrict__ hp,
    const int* __restrict__ src, const int* __restrict__ dst,
    const float* __restrict__ ah, const float* __restrict__ dt,
    const float* __restrict__ Wh /* 897 x 3 */, const float* __restrict__ bh,
    const float* __restrict__ ex, const float* __restrict__ denom,
    float* __restrict__ out, int EC)
{
  int e = (blockIdx.x * 256 + threadIdx.x) >> 5;
  int lane = threadIdx.x & 31;
  if (e >= EC) return;
  int s = src[e], d = dst[e];
  const float* hs_ = hp + (long)s * 64;
  const float* hd_ = hp + (long)d * 64;
  int c0 = lane, c1 = lane + 32;
  float xs0 = hs_[c0], xs1 = hs_[c1], xd0 = hd_[c0], xd1 = hd_[c1];
  float p0 = xs0 * Wh[c0 * 3 + 0] + xs1 * Wh[c1 * 3 + 0]
           + xd0 * Wh[(64 + c0) * 3 + 0] + xd1 * Wh[(64 + c1) * 3 + 0];
  float p1 = xs0 * Wh[c0 * 3 + 1] + xs1 * Wh[c1 * 3 + 1]
           + xd0 * Wh[(64 + c0) * 3 + 1] + xd1 * Wh[(64 + c1) * 3 + 1];
  float p2 = xs0 * Wh[c0 * 3 + 2] + xs1 * Wh[c1 * 3 + 2]
           + xd0 * Wh[(64 + c0) * 3 + 2] + xd1 * Wh[(64 + c1) * 3 + 2];
  for (int off = 16; off; off >>= 1) {
    p0 += __shfl_xor(p0, off);
    p1 += __shfl_xor(p1, off);
    p2 += __shfl_xor(p2, off);
  }
  if (lane == 0) {
    float dtv = dt[e];
    out[(long)e * 3 + 0] = p0 + ah[(long)e * 3 + 0] + dtv * Wh[896 * 3 + 0] + bh[0];
    out[(long)e * 3 + 1] = p1 + ah[(long)e * 3 + 1] + dtv * Wh[896 * 3 + 1] + bh[1];
    out[(long)e * 3 + 2] = p2 + ah[(long)e * 3 + 2] + dtv * Wh[896 * 3 + 2] + bh[2];
    out[(long)EC * 3 + e] = ex[e] / (denom[d] + 1e-16f);
  }
}

extern "C" void kernel_launch(void* const* d_in, const int* in_sizes, int n_in,
                              void* d_out, int out_size, void* d_ws, size_t ws_size,
                              hipStream_t stream) {
  const float* x_paper = (const float*)d_in[0];
  const float* x_author = (const float*)d_in[1];
  const int*   cei      = (const int*)d_in[2];
  const int*   wsrc     = (const int*)d_in[3];
  const int*   wdst     = (const int*)d_in[4];
  const float* attr     = (const float*)d_in[5];
  const float* dt       = (const float*)d_in[6];
  const float* W_pp = (const float*)d_in[7],  *b_pp = (const float*)d_in[8];
  const float* W_ap = (const float*)d_in[9],  *b_ap = (const float*)d_in[10];
  const float* Wq = (const float*)d_in[11], *bq = (const float*)d_in[12];
  const float* Wk = (const float*)d_in[13], *bk = (const float*)d_in[14];
  const float* Wv = (const float*)d_in[15], *bv = (const float*)d_in[16];
  const float* We = (const float*)d_in[17], *be = (const float*)d_in[18];
  const float* lam = (const float*)d_in[19];
  const float* sWl = (const float*)d_in[20], *sbl = (const float*)d_in[21];
  const float* sWr = (const float*)d_in[22];
  const float* Wh = (const float*)d_in[23], *bh = (const float*)d_in[24];
  float* ws = (float*)d_ws;
  float* out = (float*)d_out;

  const int* src = cei;            // row 0
  const int* dst = cei + EC_;      // row 1

  float* xp   = ws + OFF_XP;   float* xa    = ws + OFF_XA;
  float* q    = ws + OFF_Q;    float* k     = ws + OFF_K;
  float* v    = ws + OFF_V;    float* sr    = ws + OFF_SR;
  float* hs   = ws + OFF_HS;   float* ef    = ws + OFF_EF;
  float* ah   = ws + OFF_AH;   float* araw  = ws + OFF_ARAW;
  float* ex   = ws + OFF_EX;   float* mmax  = ws + OFF_MMAX;
  float* agg  = ws + OFF_AGG;  float* deg   = ws + OFF_DEG;
  float* denom= ws + OFF_DENOM;float* hgat  = ws + OFF_HGAT;
  float* hp   = ws + OFF_HP;

  // 0) init zero region + -inf segment-max seeds
  {
    long n = ZERO_LEN;
    int blocks = (int)((n + 255) / 256);
    init_ws_k<<<blocks, 256, 0, stream>>>(agg, n, mmax, NP_);
  }
  // 1) node projections
  gemm_n64_k<<<(NP_/16 + 3)/4, 128, 0, stream>>>(x_paper, W_pp, b_pp, xp, NP_/16, PD_, 1);
  gemm_n64_k<<<(NA_/16 + 3)/4, 128, 0, stream>>>(x_author, W_ap, b_ap, xa, NA_/16, AD_, 1);
  // 2) q, k, v, sage_r  (xp is L2-resident: 12.8 MB)
  gemm_n64_k<<<(NP_/16 + 3)/4, 128, 0, stream>>>(xp, Wq, bq, q, NP_/16, HD_, 0);
  gemm_n64_k<<<(NP_/16 + 3)/4, 128, 0, stream>>>(xp, Wk, bk, k, NP_/16, HD_, 0);
  gemm_n64_k<<<(NP_/16 + 3)/4, 128, 0, stream>>>(xp, Wv, bv, v, NP_/16, HD_, 0);
  gemm_n64_k<<<(NP_/16 + 3)/4, 128, 0, stream>>>(xp, sWr, nullptr, sr, NP_/16, HD_, 0);
  // 3) SAGE mean aggregation (author -> paper)
  {
    long total = (long)EW_ * 64;
    sage_scatter_k<<<(int)((total + 255)/256), 256, 0, stream>>>(xa, wsrc, wdst, agg, deg, total);
    sage_mean_k<<<(int)((NP64 + 255)/256), 256, 0, stream>>>(agg, deg, NP64);
  }
  gemm_n64_k<<<(NP_/16 + 3)/4, 128, 0, stream>>>(agg, sWl, sbl, hs, NP_/16, HD_, 0);
  // 4) fused e_feat GEMM + attr contribution to the head (single async-pipelined
  //    pass over 922 MB); dynamic LDS: 4 waves * 16 rows * ROWPAD floats
  {
    size_t lds_bytes = (size_t)4 * 16 * ROWPAD * sizeof(float);
    efeat_head_k<<<(EC_/16 + 3)/4, 128, lds_bytes, stream>>>(attr, We, be, Wh + 128 * 3, ef, ah, EC_/16);
  }
  // 5) attention raw scores + segment max
  alpha_raw_k<<<(EC_ + 7)/8, 256, 0, stream>>>(q, k, ef, src, dst, dt, lam, araw, mmax, EC_);
  // 6) exp + denom + weighted V scatter
  ex_scatter_k<<<(EC_ + 7)/8, 256, 0, stream>>>(v, araw, mmax, src, dst, ex, denom, hgat, EC_);
  // 7) h_papers = relu(h_sage + h_gat)
  finalize_k<<<(int)((NP64 + 255)/256), 256, 0, stream>>>(hs, sr, hgat, denom, hp, NP64);
  // 8) intent head + normalized alpha output
  head_k<<<(EC_ + 7)/8, 256, 0, stream>>>(hp, src, dst, ah, dt, Wh, bh, ex, denom, out, EC_);
}